// GAT_85950885528245
// MI455X (gfx1250) — compile-verified
//
#include <hip/hip_runtime.h>

// ---------------------------------------------------------------------------
// GATv2Conv (heads=1, edge_dim=1) + global_mean_pool for MI455X (gfx1250).
// Node transforms use fp32 WMMA (V_WMMA_F32_16X16X4_F32, wave32).
// ---------------------------------------------------------------------------

#define N_NODES  100000
#define N_EDGES  3200000
#define F_IN     128
#define F_OUT    32
#define N_GRAPHS 64

typedef __attribute__((ext_vector_type(2))) float v2f;
typedef __attribute__((ext_vector_type(8))) float v8f;

// Workspace layout (in floats). Total ~52 MB.
#define WS_XL     0                        // N*32
#define WS_XR     (WS_XL  + N_NODES*F_OUT) // N*32
#define WS_OUT    (WS_XR  + N_NODES*F_OUT) // N*32
#define WS_LOGIT  (WS_OUT + N_NODES*F_OUT) // E  (reused as alpha in pass 2/3)
#define WS_MX     (WS_LOGIT + N_EDGES)     // N  (uint-encoded running max)
#define WS_DENOM  (WS_MX  + N_NODES)       // N
#define WS_POOL   (WS_DENOM + N_NODES)     // 64*32
#define WS_CNT    (WS_POOL + N_GRAPHS*F_OUT) // 64

// Monotone order-preserving float <-> uint map (for atomic max over signed floats)
__device__ __forceinline__ unsigned fenc(float f) {
  unsigned u = __float_as_uint(f);
  return (u & 0x80000000u) ? ~u : (u | 0x80000000u);
}
__device__ __forceinline__ float fdec(unsigned u) {
  unsigned b = (u & 0x80000000u) ? (u & 0x7FFFFFFFu) : ~u;
  return __uint_as_float(b);
}
__device__ __forceinline__ float lrelu(float v) { return fmaxf(v, 0.2f * v); }

// ---------------------------------------------------------------------------
// 0) init: zero accumulators, seed per-node max with encoded(-inf)
// ---------------------------------------------------------------------------
__global__ void gat_init(float* __restrict__ ws) {
  int idx = blockIdx.x * blockDim.x + threadIdx.x;   // covers N_NODES*F_OUT
  if (idx < N_NODES * F_OUT) ws[WS_OUT + idx] = 0.0f;
  if (idx < N_NODES) {
    ws[WS_DENOM + idx] = 0.0f;
    ((unsigned*)ws)[WS_MX + idx] = 0x007FFFFFu;      // fenc(-inf)
  }
  if (idx < N_GRAPHS * F_OUT) ws[WS_POOL + idx] = 0.0f;
  if (idx < N_GRAPHS) ws[WS_CNT + idx] = 0.0f;
}

// ---------------------------------------------------------------------------
// 1) Node transform: xl = x@W_l + b_l, xr = x@W_r + b_r via fp32 WMMA.
//    One wave32 computes one 16-row x 32-col tile for BOTH weight matrices.
//    A (16x4 f32): lane<16 holds row=lane, K={kk,kk+1}; lane>=16 K={kk+2,kk+3}
//    B (4x16 f32): lane<16 holds col=lane, K={kk,kk+1}; lane>=16 K={kk+2,kk+3}
//    D (16x16 f32): VGPR r, lane<16 -> M=r, lane>=16 -> M=8+r, N=lane&15
// ---------------------------------------------------------------------------
__global__ void gat_node_transform(const float* __restrict__ x,
                                   const float* __restrict__ Wl,
                                   const float* __restrict__ bl,
                                   const float* __restrict__ Wr,
                                   const float* __restrict__ br,
                                   float* __restrict__ xl,
                                   float* __restrict__ xr) {
  const int wave = (blockIdx.x * blockDim.x + threadIdx.x) >> 5;
  const int lane = threadIdx.x & 31;
  if (wave >= N_NODES / 16) return;        // uniform per wave: EXEC all-1 at WMMA
  const int row0  = wave * 16;
  const int m     = lane & 15;             // row (A) / col (B) within tile
  const int khalf = (lane >> 4) << 1;      // 0 or 2

  v8f accL0 = {}, accL1 = {}, accR0 = {}, accR1 = {};
  const float* xrow = x + (size_t)(row0 + m) * F_IN + khalf;

  #pragma unroll 4
  for (int kk = 0; kk < F_IN; kk += 4) {
    v2f a;
    a.x = xrow[kk];
    a.y = xrow[kk + 1];
    const int kr = kk + khalf;
    v2f bL0, bL1, bR0, bR1;
    bL0.x = Wl[kr * F_OUT + m];        bL0.y = Wl[(kr + 1) * F_OUT + m];
    bL1.x = Wl[kr * F_OUT + 16 + m];   bL1.y = Wl[(kr + 1) * F_OUT + 16 + m];
    bR0.x = Wr[kr * F_OUT + m];        bR0.y = Wr[(kr + 1) * F_OUT + m];
    bR1.x = Wr[kr * F_OUT + 16 + m];   bR1.y = Wr[(kr + 1) * F_OUT + 16 + m];
    accL0 = __builtin_amdgcn_wmma_f32_16x16x4_f32(false, a, false, bL0, (short)0, accL0, false, false);
    accL1 = __builtin_amdgcn_wmma_f32_16x16x4_f32(false, a, false, bL1, (short)0, accL1, false, false);
    accR0 = __builtin_amdgcn_wmma_f32_16x16x4_f32(false, a, false, bR0, (short)0, accR0, false, false);
    accR1 = __builtin_amdgcn_wmma_f32_16x16x4_f32(false, a, false, bR1, (short)0, accR1, false, false);
  }

  const int rowoff = (lane >> 4) * 8;
  const int col    = lane & 15;
  #pragma unroll
  for (int r = 0; r < 8; ++r) {
    const size_t row = (size_t)(row0 + rowoff + r);
    xl[row * F_OUT + col]      = accL0[r] + bl[col];
    xl[row * F_OUT + 16 + col] = accL1[r] + bl[16 + col];
    xr[row * F_OUT + col]      = accR0[r] + br[col];
    xr[row * F_OUT + 16 + col] = accR1[r] + br[16 + col];
  }
}

// ---------------------------------------------------------------------------
// 2) Edge pass 1: logit = leaky_relu(xl[src]+xr[dst]+ea*W_e) . att ; seg-max
// ---------------------------------------------------------------------------
__global__ void gat_edge_logit(const int* __restrict__ src,
                               const int* __restrict__ dst,
                               const float* __restrict__ edge_attr,
                               const float* __restrict__ We,
                               const float* __restrict__ att,
                               const float* __restrict__ xl,
                               const float* __restrict__ xr,
                               float* __restrict__ logit,
                               unsigned* __restrict__ mxenc) {
  const int e = blockIdx.x * blockDim.x + threadIdx.x;
  if (e >= N_EDGES) return;
  const int s = src[e];
  const int d = dst[e];
  const float ea = edge_attr[e];
  const float4* a4 = (const float4*)(xl + (size_t)s * F_OUT);
  const float4* b4 = (const float4*)(xr + (size_t)d * F_OUT);
  const float4* w4 = (const float4*)We;
  const float4* t4 = (const float4*)att;
  float acc = 0.0f;
  #pragma unroll
  for (int i = 0; i < F_OUT / 4; ++i) {
    const float4 a = a4[i], b = b4[i], w = w4[i], t = t4[i];
    acc = fmaf(lrelu(fmaf(ea, w.x, a.x + b.x)), t.x, acc);
    acc = fmaf(lrelu(fmaf(ea, w.y, a.y + b.y)), t.y, acc);
    acc = fmaf(lrelu(fmaf(ea, w.z, a.z + b.z)), t.z, acc);
    acc = fmaf(lrelu(fmaf(ea, w.w, a.w + b.w)), t.w, acc);
  }
  logit[e] = acc;
  atomicMax(mxenc + d, fenc(acc));
}

// ---------------------------------------------------------------------------
// 3) Edge pass 2: alpha = exp(logit - mx[dst]) (in place); seg-sum denom
// ---------------------------------------------------------------------------
__global__ void gat_edge_alpha(const int* __restrict__ dst,
                               const unsigned* __restrict__ mxenc,
                               float* __restrict__ logit_alpha,
                               float* __restrict__ denom) {
  const int e = blockIdx.x * blockDim.x + threadIdx.x;
  if (e >= N_EDGES) return;
  const int d = dst[e];
  const float a = __expf(logit_alpha[e] - fdec(mxenc[d]));
  logit_alpha[e] = a;
  atomicAdd(denom + d, a);
}

// ---------------------------------------------------------------------------
// 4) Edge pass 3: out[dst] += (alpha/denom[dst]) * xl[src]   (f32 atomics, L2)
// ---------------------------------------------------------------------------
__global__ void gat_edge_aggregate(const int* __restrict__ src,
                                   const int* __restrict__ dst,
                                   const float* __restrict__ alpha,
                                   const float* __restrict__ denom,
                                   const float* __restrict__ xl,
                                   float* __restrict__ out) {
  const int e = blockIdx.x * blockDim.x + threadIdx.x;
  if (e >= N_EDGES) return;
  const int s = src[e];
  const int d = dst[e];
  const float w = alpha[e] / (denom[d] + 1e-16f);
  const float4* a4 = (const float4*)(xl + (size_t)s * F_OUT);
  float* o = out + (size_t)d * F_OUT;
  #pragma unroll
  for (int i = 0; i < F_OUT / 4; ++i) {
    const float4 a = a4[i];
    atomicAdd(o + 4 * i + 0, w * a.x);
    atomicAdd(o + 4 * i + 1, w * a.y);
    atomicAdd(o + 4 * i + 2, w * a.z);
    atomicAdd(o + 4 * i + 3, w * a.w);
  }
}

// ---------------------------------------------------------------------------
// 5) Pool scatter: pooled[batch[n]] += out[n] + bias ; cnt[batch[n]] += 1
// ---------------------------------------------------------------------------
__global__ void gat_pool(const int* __restrict__ batch,
                         const float* __restrict__ bias,
                         const float* __restrict__ out,
                         float* __restrict__ pooled,
                         float* __restrict__ cnt) {
  const int idx = blockIdx.x * blockDim.x + threadIdx.x;  // N_NODES*F_OUT
  if (idx >= N_NODES * F_OUT) return;
  const int n = idx >> 5;
  const int f = idx & 31;
  const int g = batch[n];
  atomicAdd(pooled + g * F_OUT + f, out[idx] + bias[f]);
  if (f == 0) atomicAdd(cnt + g, 1.0f);
}

// ---------------------------------------------------------------------------
// 6) Finalize: d_out = pooled / max(cnt, 1)
// ---------------------------------------------------------------------------
__global__ void gat_finalize(const float* __restrict__ pooled,
                             const float* __restrict__ cnt,
                             float* __restrict__ dout) {
  const int idx = blockIdx.x * blockDim.x + threadIdx.x;  // N_GRAPHS*F_OUT
  if (idx >= N_GRAPHS * F_OUT) return;
  dout[idx] = pooled[idx] / fmaxf(cnt[idx >> 5], 1.0f);
}

// ---------------------------------------------------------------------------
extern "C" void kernel_launch(void* const* d_in, const int* in_sizes, int n_in,
                              void* d_out, int out_size, void* d_ws, size_t ws_size,
                              hipStream_t stream) {
  const float* x         = (const float*)d_in[0];
  const float* edge_attr = (const float*)d_in[1];
  const float* W_l       = (const float*)d_in[2];
  const float* b_l       = (const float*)d_in[3];
  const float* W_r       = (const float*)d_in[4];
  const float* b_r       = (const float*)d_in[5];
  const float* W_e       = (const float*)d_in[6];
  const float* att       = (const float*)d_in[7];
  const float* bias      = (const float*)d_in[8];
  const int*   eidx      = (const int*)d_in[9];   // [2, E]: row 0 = src, row 1 = dst
  const int*   batch     = (const int*)d_in[10];

  const int* src = eidx;
  const int* dst = eidx + N_EDGES;

  float*    ws    = (float*)d_ws;
  float*    xl    = ws + WS_XL;
  float*    xr    = ws + WS_XR;
  float*    outv  = ws + WS_OUT;
  float*    logit = ws + WS_LOGIT;   // reused as alpha
  unsigned* mxenc = ((unsigned*)ws) + WS_MX;
  float*    denom = ws + WS_DENOM;
  float*    pool  = ws + WS_POOL;
  float*    cnt   = ws + WS_CNT;

  const int BT = 256;
  const int gridNF  = (N_NODES * F_OUT + BT - 1) / BT;   // 12500
  const int gridE   = (N_EDGES + BT - 1) / BT;           // 12500
  const int gridMM  = ((N_NODES / 16) + (BT / 32) - 1) / (BT / 32); // 782 (8 waves/blk)
  const int gridFin = (N_GRAPHS * F_OUT + BT - 1) / BT;  // 8

  gat_init<<<gridNF, BT, 0, stream>>>(ws);
  gat_node_transform<<<gridMM, BT, 0, stream>>>(x, W_l, b_l, W_r, b_r, xl, xr);
  gat_edge_logit<<<gridE, BT, 0, stream>>>(src, dst, edge_attr, W_e, att, xl, xr,
                                           logit, mxenc);
  gat_edge_alpha<<<gridE, BT, 0, stream>>>(dst, mxenc, logit, denom);
  gat_edge_aggregate<<<gridE, BT, 0, stream>>>(src, dst, logit, denom, xl, outv);
  gat_pool<<<gridNF, BT, 0, stream>>>(batch, bias, outv, pool, cnt);
  gat_finalize<<<gridFin, BT, 0, stream>>>(pool, cnt, (float*)d_out);
}